// SmalldeckClassification_37434934952727
// MI455X (gfx1250) — compile-verified
//
#include <hip/hip_runtime.h>
#include <hip/hip_bf16.h>

typedef __attribute__((ext_vector_type(2))) float v2f;
typedef __attribute__((ext_vector_type(8))) float v8f;

#define MROWS   512
#define NCOMBO  60
#define NCLS    7463
#define KH      512
#define RAWF    15360   // 60*256
#define FLT_BIG 3.402823466e+38f

// ---------------------------------------------------------------------------
// Combo tables: C([0..3],2) x C([4..8],3), lexicographic (itertools order)
// ---------------------------------------------------------------------------
__device__ __constant__ int cH0[6]  = {0,0,0,1,1,2};
__device__ __constant__ int cH1[6]  = {1,2,3,2,3,3};
__device__ __constant__ int cB0[10] = {4,4,4,4,4,4,5,5,5,6};
__device__ __constant__ int cB1[10] = {5,5,5,6,6,7,6,6,7,7};
__device__ __constant__ int cB2[10] = {6,7,8,7,8,8,7,8,8,8};

__device__ inline void stable_sort_cards(int* r, int* s, int n) {
  // sort ascending by (rank, suit), stable => matches double stable argsort
  for (int i = 1; i < n; ++i) {
    int kr = r[i], ks = s[i], key = kr * 8 + ks;
    int j = i - 1;
    while (j >= 0 && (r[j] * 8 + s[j]) > key) {
      r[j + 1] = r[j]; s[j + 1] = s[j]; --j;
    }
    r[j + 1] = kr; s[j + 1] = ks;
  }
}

// ---------------------------------------------------------------------------
// Kernel 1: unspool + feature nets + BN/ReLU + h1/h2 MLP
//   writes raw_flat (512 x 15360) and h512 (30720 x 512); seeds best[m]
// ---------------------------------------------------------------------------
__global__ __launch_bounds__(256) void feat_kernel(
    const int* __restrict__ x,
    const float* __restrict__ w_suit, const float* __restrict__ b_suit,
    const float* __restrict__ g_suit, const float* __restrict__ be_suit,
    const float* __restrict__ w_rank, const float* __restrict__ b_rank,
    const float* __restrict__ g_rank, const float* __restrict__ be_rank,
    const float* __restrict__ w_h1,   const float* __restrict__ b_h1,
    const float* __restrict__ w_h2,   const float* __restrict__ b_h2,
    float* __restrict__ raw_flat, float* __restrict__ h512,
    int* __restrict__ best_ws)
{
  __shared__ float outb[NCOMBO * 256];  // (n, o, slot) pre/post activations
  __shared__ float tmp[512];            // BN partials, then h1 scratch
  __shared__ float s_mu[32], s_sc[32], s_bt[32];
  __shared__ int   s_rank[9], s_suit[9];

  const int m   = blockIdx.x;
  const int tid = threadIdx.x;

  if (tid == 0) {
    best_ws[m] = 0x7FFFFFFF;
    int r[9], s[9];
    for (int c = 0; c < 9; ++c) {
      r[c] = x[m * 18 + 2 * c];
      s[c] = x[m * 18 + 2 * c + 1];
    }
    stable_sort_cards(r, s, 4);          // hole
    stable_sort_cards(r + 4, s + 4, 5);  // board
    for (int c = 0; c < 9; ++c) { s_rank[c] = r[c]; s_suit[c] = s[c]; }
  }
  __syncthreads();

  // ---- pre-activations: slot 0..10 = rank conv, slot 11..15 = suit ----
  for (int idx = tid; idx < NCOMBO * 256; idx += 256) {
    const int n    = idx >> 8;
    const int rem  = idx & 255;
    const int o    = rem >> 4;
    const int slot = rem & 15;
    const int hp = n / 10, bi = n % 10;
    int card[5];
    card[0] = cH0[hp]; card[1] = cH1[hp];
    card[2] = cB0[bi]; card[3] = cB1[bi]; card[4] = cB2[bi];
    float val;
    if (slot < 11) {
      const int t = slot;
      val = b_rank[o];
      #pragma unroll
      for (int c = 0; c < 5; ++c) {
        const int h = s_rank[card[c]] - t;
        if (h >= 0 && h < 5) val += w_rank[o * 25 + c * 5 + h];
      }
    } else {
      const int l = slot - 11;
      val = b_suit[o];
      #pragma unroll
      for (int c = 0; c < 5; ++c)
        if (s_suit[card[c]] == l) val += w_suit[o * 5 + c];
    }
    outb[idx] = val;
  }
  __syncthreads();

  // ---- BN stats: 32 groups = 16 o x {rank(660 elems), suit(300 elems)} ----
  {
    const int g = tid >> 3, part = tid & 7;
    const int o = g >> 1, kind = g & 1;
    const int cnt = kind ? 300 : 660;
    float sum = 0.f, sq = 0.f;
    for (int e = part; e < cnt; e += 8) {
      int idx;
      if (kind) { int n = e / 5,  l = e % 5;  idx = n * 256 + o * 16 + 11 + l; }
      else      { int n = e / 11, t = e % 11; idx = n * 256 + o * 16 + t; }
      const float v = outb[idx];
      sum += v; sq += v * v;
    }
    tmp[g * 8 + part]       = sum;
    tmp[256 + g * 8 + part] = sq;
  }
  __syncthreads();
  if (tid < 32) {
    float S = 0.f, Q = 0.f;
    for (int p = 0; p < 8; ++p) { S += tmp[tid * 8 + p]; Q += tmp[256 + tid * 8 + p]; }
    const int o = tid >> 1, kind = tid & 1;
    const float cntf = kind ? 300.f : 660.f;
    const float mu   = S / cntf;
    const float var  = Q / cntf - mu * mu;
    const float gam  = kind ? g_suit[o]  : g_rank[o];
    const float bet  = kind ? be_suit[o] : be_rank[o];
    s_mu[tid] = mu;
    s_sc[tid] = gam * rsqrtf(var + 1e-5f);
    s_bt[tid] = bet;
  }
  __syncthreads();

  // ---- normalize + ReLU, emit raw_flat ----
  for (int idx = tid; idx < NCOMBO * 256; idx += 256) {
    const int rem = idx & 255;
    const int o = rem >> 4, slot = rem & 15;
    const int g = o * 2 + (slot >= 11 ? 1 : 0);
    float v = (outb[idx] - s_mu[g]) * s_sc[g] + s_bt[g];
    v = v > 0.f ? v : 0.f;
    outb[idx] = v;
    raw_flat[(size_t)m * RAWF + idx] = v;
  }
  __syncthreads();

  // ---- h1 (16->32) then h2 (32->32) per (n,o); h512 row = m*60+n ----
  for (int n = 0; n < NCOMBO; ++n) {
    for (int q = tid; q < 512; q += 256) {
      const int o = q >> 5, j = q & 31;
      float acc = b_h1[j];
      const float* src = &outb[n * 256 + o * 16];
      #pragma unroll
      for (int i = 0; i < 16; ++i) acc += src[i] * w_h1[i * 32 + j];
      tmp[q] = acc > 0.f ? acc : 0.f;
    }
    __syncthreads();
    for (int q = tid; q < 512; q += 256) {
      const int o = q >> 5, j = q & 31;
      float acc = b_h2[j];
      const float* src = &tmp[o * 32];
      #pragma unroll
      for (int i = 0; i < 32; ++i) acc += src[i] * w_h2[i * 32 + j];
      acc = acc > 0.f ? acc : 0.f;
      h512[((size_t)m * NCOMBO + n) * KH + q] = acc;
    }
    __syncthreads();
  }
}

// ---------------------------------------------------------------------------
// Kernel 2: logits = h512(30720x512) * w_cat(512x7463), fp32 WMMA 16x16x4,
//           fused argmax over classes + atomicMin over combos.
//   Pad columns are handled by CLAMPING the column index (no predicated
//   loads -> no exec divergence in the inner K loop); the argmax update
//   (once per 128-WMMA tile) masks the invalid lanes exactly.
// ---------------------------------------------------------------------------
__global__ __launch_bounds__(256) void logits_argmax_kernel(
    const float* __restrict__ h512, const float* __restrict__ w_cat,
    const float* __restrict__ b_cat, int* __restrict__ best_ws)
{
  __shared__ float Asm[16 * 516];   // 516 pitch -> conflict-free M reads
  __shared__ float redV[8 * 16];
  __shared__ int   redC[8 * 16];

  const int row0 = blockIdx.x * 16;
  const int tid  = threadIdx.x;

  for (int idx = tid; idx < 16 * 512; idx += 256) {
    const int rr = idx >> 9, kk = idx & 511;
    Asm[rr * 516 + kk] = h512[(size_t)(row0 + rr) * KH + kk];
  }
  __syncthreads();

  const int wave  = tid >> 5;
  const int lane  = tid & 31;
  const int laneN = lane & 15;
  const int hi    = lane >> 4;   // 0: K{0,1}/rows 0-7 of C, 1: K{2,3}/rows 8-15

  float bestV[8];
  int   bestC[8];
  #pragma unroll
  for (int j = 0; j < 8; ++j) { bestV[j] = -FLT_BIG; bestC[j] = 0x7FFFFFFF; }

  const float* arow = &Asm[laneN * 516 + hi * 2];
  const int iters = (NCLS + 127) / 128;   // 59
  for (int it = 0; it < iters; ++it) {
    const int col   = it * 128 + wave * 16 + laneN;
    const int colc  = col < NCLS ? col : (NCLS - 1);   // clamp: loads stay unconditional
    const float* bcol = w_cat + (size_t)(hi * 2) * NCLS + colc;
    v8f c = {};
    #pragma unroll 8
    for (int k0 = 0; k0 < KH; k0 += 4) {
      v2f a, b;
      a.x = arow[k0];
      a.y = arow[k0 + 1];
      b.x = bcol[(size_t)k0 * NCLS];
      b.y = bcol[(size_t)(k0 + 1) * NCLS];
      c = __builtin_amdgcn_wmma_f32_16x16x4_f32(false, a, false, b,
                                                (short)0, c, false, false);
    }
    if (col < NCLS) {
      const float bias = b_cat[col];
      #pragma unroll
      for (int j = 0; j < 8; ++j) {
        const float v = c[j] + bias;
        if (v > bestV[j]) { bestV[j] = v; bestC[j] = col; }  // cols ascend => first-max kept
      }
    }
  }

  // reduce across the 16 lanes of each half (rows live in lane halves)
  #pragma unroll
  for (int j = 0; j < 8; ++j) {
    float v = bestV[j]; int ci = bestC[j];
    for (int d = 1; d < 16; d <<= 1) {
      const float ov = __shfl_xor(v, d, 32);
      const int   oc = __shfl_xor(ci, d, 32);
      if (ov > v || (ov == v && oc < ci)) { v = ov; ci = oc; }
    }
    bestV[j] = v; bestC[j] = ci;
  }
  if (laneN == 0) {
    #pragma unroll
    for (int j = 0; j < 8; ++j) {
      const int row = hi * 8 + j;
      redV[wave * 16 + row] = bestV[j];
      redC[wave * 16 + row] = bestC[j];
    }
  }
  __syncthreads();
  if (tid < 16) {
    const int row = tid;
    float v = redV[row]; int ci = redC[row];
    for (int w = 1; w < 8; ++w) {
      const float ov = redV[w * 16 + row];
      const int   oc = redC[w * 16 + row];
      if (ov > v || (ov == v && oc < ci)) { v = ov; ci = oc; }
    }
    const int r = row0 + row;
    atomicMin(&best_ws[r / NCOMBO], ci);   // min over the 60 combos
  }
}

// ---------------------------------------------------------------------------
// Kernel 3: rr1 = relu(raw_flat(512x15360) * w_o1(15360x512) + b_o1), WMMA
// ---------------------------------------------------------------------------
__global__ __launch_bounds__(256) void rr1_kernel(
    const float* __restrict__ X, const float* __restrict__ w_o1,
    const float* __restrict__ b_o1, float* __restrict__ rr1)
{
  __shared__ float Asm[16 * 516];
  const int row0 = blockIdx.x * 16;
  const int tid  = threadIdx.x;
  const int wave = tid >> 5, lane = tid & 31;
  const int laneN = lane & 15, hi = lane >> 4;
  const int col = blockIdx.y * 128 + wave * 16 + laneN;

  v8f c = {};
  for (int kc = 0; kc < RAWF; kc += 512) {
    __syncthreads();
    for (int idx = tid; idx < 16 * 512; idx += 256) {
      const int rr = idx >> 9, kk = idx & 511;
      Asm[rr * 516 + kk] = X[(size_t)(row0 + rr) * RAWF + kc + kk];
    }
    __syncthreads();
    const float* arow = &Asm[laneN * 516 + hi * 2];
    const float* bcol = w_o1 + (size_t)(kc + hi * 2) * 512 + col;
    #pragma unroll 8
    for (int k0 = 0; k0 < 512; k0 += 4) {
      v2f a, b;
      a.x = arow[k0];
      a.y = arow[k0 + 1];
      b.x = bcol[(size_t)k0 * 512];
      b.y = bcol[(size_t)(k0 + 1) * 512];
      c = __builtin_amdgcn_wmma_f32_16x16x4_f32(false, a, false, b,
                                                (short)0, c, false, false);
    }
  }
  const float bias = b_o1[col];
  #pragma unroll
  for (int j = 0; j < 8; ++j) {
    float v = c[j] + bias;
    v = v > 0.f ? v : 0.f;
    rr1[(size_t)(row0 + hi * 8 + j) * 512 + col] = v;
  }
}

// ---------------------------------------------------------------------------
// Kernel 4: tail MLP 512->256->127, concat class, x w_sc -> (512,6)
// ---------------------------------------------------------------------------
__global__ __launch_bounds__(256) void tail_kernel(
    const float* __restrict__ rr1,
    const float* __restrict__ w_o2, const float* __restrict__ b_o2,
    const float* __restrict__ w_o3, const float* __restrict__ b_o3,
    const float* __restrict__ w_sc, const float* __restrict__ b_sc,
    const int* __restrict__ best_ws, float* __restrict__ out)
{
  __shared__ float a[512], r2[256], r3[128];
  const int m = blockIdx.x, tid = threadIdx.x;
  for (int i = tid; i < 512; i += 256) a[i] = rr1[(size_t)m * 512 + i];
  __syncthreads();
  {
    float acc = b_o2[tid];
    for (int i = 0; i < 512; ++i) acc += a[i] * w_o2[i * 256 + tid];
    r2[tid] = acc > 0.f ? acc : 0.f;
  }
  __syncthreads();
  if (tid < 127) {
    float acc = b_o3[tid];
    for (int i = 0; i < 256; ++i) acc += r2[i] * w_o3[i * 127 + tid];
    r3[tid] = acc > 0.f ? acc : 0.f;
  }
  if (tid == 0) r3[127] = (float)best_ws[m];
  __syncthreads();
  if (tid < 6) {
    float acc = b_sc[tid];
    for (int i = 0; i < 128; ++i) acc += r3[i] * w_sc[i * 6 + tid];
    out[m * 6 + tid] = acc;
  }
}

// ---------------------------------------------------------------------------
extern "C" void kernel_launch(void* const* d_in, const int* in_sizes, int n_in,
                              void* d_out, int out_size, void* d_ws, size_t ws_size,
                              hipStream_t stream) {
  const int*   x       = (const int*)  d_in[0];
  const float* w_suit  = (const float*)d_in[1];
  const float* b_suit  = (const float*)d_in[2];
  const float* g_suit  = (const float*)d_in[3];
  const float* be_suit = (const float*)d_in[4];
  const float* w_rank  = (const float*)d_in[5];
  const float* b_rank  = (const float*)d_in[6];
  const float* g_rank  = (const float*)d_in[7];
  const float* be_rank = (const float*)d_in[8];
  const float* w_h1    = (const float*)d_in[9];
  const float* b_h1    = (const float*)d_in[10];
  const float* w_h2    = (const float*)d_in[11];
  const float* b_h2    = (const float*)d_in[12];
  const float* w_cat   = (const float*)d_in[13];
  const float* b_cat   = (const float*)d_in[14];
  const float* w_o1    = (const float*)d_in[15];
  const float* b_o1    = (const float*)d_in[16];
  const float* w_o2    = (const float*)d_in[17];
  const float* b_o2    = (const float*)d_in[18];
  const float* w_o3    = (const float*)d_in[19];
  const float* b_o3    = (const float*)d_in[20];
  const float* w_sc    = (const float*)d_in[21];
  const float* b_sc    = (const float*)d_in[22];

  float* raw_flat = (float*)d_ws;                                  // 512*15360
  float* h512     = raw_flat + (size_t)MROWS * RAWF;               // 30720*512
  float* rr1      = h512 + (size_t)MROWS * NCOMBO * KH;            // 512*512
  int*   best     = (int*)(rr1 + (size_t)MROWS * 512);             // 512
  float* out      = (float*)d_out;

  feat_kernel<<<MROWS, 256, 0, stream>>>(
      x, w_suit, b_suit, g_suit, be_suit, w_rank, b_rank, g_rank, be_rank,
      w_h1, b_h1, w_h2, b_h2, raw_flat, h512, best);

  logits_argmax_kernel<<<(MROWS * NCOMBO) / 16, 256, 0, stream>>>(
      h512, w_cat, b_cat, best);

  rr1_kernel<<<dim3(MROWS / 16, 4), 256, 0, stream>>>(raw_flat, w_o1, b_o1, rr1);

  tail_kernel<<<MROWS, 256, 0, stream>>>(
      rr1, w_o2, b_o2, w_o3, b_o3, w_sc, b_sc, best, out);
}